// ROIGatherBlock_11948599017610
// MI455X (gfx1250) — compile-verified
//
#include <hip/hip_runtime.h>
#include <hip/hip_bf16.h>
#include <math.h>

// ---------------- constants from the reference ----------------
#define NB     32          // B
#define NC     64          // C
#define FH_    80
#define FW_    200
#define PLANE  (FH_*FW_)   // 16000
#define NLANES 64
#define NSP    36
#define PE_    78
#define NR     250         // RH*RW pooled cells
#define NRP    256         // padded
#define CONVK  5
#define TOUT   32          // SP - K + 1
#define KFC    2048        // C*32
#define KG     4096        // LANES*C
#define ROIST  66          // padded LDS stride for roiT (2-word bank skew)

typedef __attribute__((ext_vector_type(2))) float v2f;
typedef __attribute__((ext_vector_type(8))) float v8f;

__device__ __forceinline__ v8f wmma4(v2f a, v2f b, v8f c) {
    // D(16x16,f32) += A(16x4,f32) x B(4x16,f32)
    return __builtin_amdgcn_wmma_f32_16x16x4_f32(false, a, false, b,
                                                 (short)0, c, false, false);
}

// ---------------------------------------------------------------
// Kernel 0: transpose conv1d_w (O,I,5) -> wr[k][o][i]  (i contiguous)
// so conv A-fragments become single aligned b64 loads.
// ---------------------------------------------------------------
__global__ void k_prep_w(const float* __restrict__ w5, float* __restrict__ wr) {
    for (int e = blockIdx.x * blockDim.x + threadIdx.x;
         e < NC * NC * CONVK; e += gridDim.x * blockDim.x) {
        int i = e & 63;
        int o = (e >> 6) & 63;
        int k = e >> 12;                 // 0..4
        wr[e] = w5[((size_t)o * NC + i) * CONVK + k];
    }
}

// ---------------------------------------------------------------
// Kernel 1: 8x8 average pool of feature map -> x_f (B,C,256 padded)
// and transposed x_fT (B,256,64). One block per (b,c) plane.
// ---------------------------------------------------------------
__global__ void k_pool(const float* __restrict__ feat,
                       float* __restrict__ xf, float* __restrict__ xfT) {
    int bc = blockIdx.x;                 // b*64 + c
    const float* plane = feat + (size_t)bc * PLANE;
    int t = threadIdx.x;                 // 0..255
    float val = 0.0f;
    if (t < NR) {
        int rh = t / 25, rw = t % 25;
        const float* p0 = plane + rh * 8 * FW_ + rw * 8;
        float s = 0.0f;
#pragma unroll
        for (int dy = 0; dy < 8; ++dy) {
            const float4* p4 = (const float4*)(p0 + dy * FW_);
            float4 u = p4[0], v = p4[1];
            s += u.x + u.y + u.z + u.w + v.x + v.y + v.z + v.w;
        }
        val = s * (1.0f / 64.0f);
    }
    int b = bc >> 6, c = bc & 63;
    xf [((size_t)b * NC  + c) * NRP + t] = val;   // pads (t>=250) get 0
    xfT[((size_t)b * NRP + t) * NC  + c] = val;
}

// ---------------------------------------------------------------
// Kernel 2: ROI 4-corner gather + conv1d(K=5) via WMMA f32 16x16x4.
// One block (256 thr = 8 waves) per (b,lane). Writes yflat[s][o*32+t].
// A frag: one global b64 from wr[k][o][i]; B frag: one LDS b64 from roiT.
// ---------------------------------------------------------------
__global__ void k_roi_conv(const float* __restrict__ feat,
                           const float* __restrict__ prior,
                           const float* __restrict__ wr,     // (5,64,64)
                           const float* __restrict__ bconv,  // (64)
                           float* __restrict__ yflat) {      // (2048,2048)
    __shared__ float roiT[NSP][ROIST];   // [t][i], padded stride 66
    __shared__ int   sidx[4][NSP];

    int blk = blockIdx.x;                // b*64 + lane
    int b = blk >> 6, ln = blk & 63;
    int tid = threadIdx.x;

    if (tid < NSP) {
        int i = 2 * tid;                 // ::stride(2) point index
        float px = prior[((size_t)b * NLANES + ln) * PE_ + 6 + i] * 0.25f;
        px = fminf(px, (float)(FW_ - 1));
        float py = fminf((319.0f - (320.0f / 71.0f) * (float)i) * 0.25f,
                         (float)(FH_ - 1));
        int xc = (int)ceilf(px),  xl = (int)floorf(px);
        int yc = (int)ceilf(py),  yl = (int)floorf(py);
        sidx[0][tid] = yc * FW_ + xc;
        sidx[1][tid] = yc * FW_ + xl;
        sidx[2][tid] = yl * FW_ + xc;
        sidx[3][tid] = yl * FW_ + xl;
    }
    __syncthreads();

    const float* fb = feat + (size_t)b * NC * PLANE;
    for (int e = tid; e < NC * NSP; e += blockDim.x) {
        int c = e / NSP, k = e % NSP;
        const float* pc = fb + (size_t)c * PLANE;
        float s = pc[sidx[0][k]] + pc[sidx[1][k]] +
                  pc[sidx[2][k]] + pc[sidx[3][k]];
        roiT[k][c] = s * 0.25f;
    }
    __syncthreads();

    // y(64x32) = bias + sum_k Wk(64x64) @ roi[:, k:k+32]
    int wave = tid >> 5;                 // 0..7
    int m0 = (wave >> 1) * 16;           // out-channel tile: 0,16,32,48
    int n0 = (wave & 1) * 16;            // t tile: 0,16
    int ml = tid & 15;
    int hi = (tid >> 4) & 1;
    int kb = hi * 2;

    v8f acc = {};
#pragma unroll 1
    for (int k = 0; k < CONVK; ++k) {
        const float* wk = wr + (size_t)k * NC * NC + (size_t)(m0 + ml) * NC + kb;
        const float* rt = &roiT[k + n0 + ml][kb];
#pragma unroll 4
        for (int i0 = 0; i0 < NC; i0 += 4) {
            v2f a  = *(const v2f*)(wk + i0);   // A[m][kb..kb+1] = Wk[o][i]
            v2f bf = *(const v2f*)(rt + i0);   // B[kb..kb+1][n] = roi[i][t]
            acc = wmma4(a, bf, acc);
        }
    }
    size_t srow = (size_t)blk * KFC;
#pragma unroll
    for (int v = 0; v < 8; ++v) {
        int o = m0 + v + hi * 8;
        int t = n0 + ml;
        yflat[srow + o * TOUT + t] = acc[v] + bconv[o];
    }
}

// ---------------------------------------------------------------
// Kernel 3: fc GEMM  x_p(2048x64) = yflat(2048x2048) @ fc_w^T + fc_b
// 512 wave-tiles; 64 blocks x 8 waves.
// ---------------------------------------------------------------
__global__ void k_fc(const float* __restrict__ yflat,
                     const float* __restrict__ fcw,   // (64,2048)
                     const float* __restrict__ fcb,   // (64)
                     float* __restrict__ xp) {        // (2048,64)
    int gw = blockIdx.x * (blockDim.x >> 5) + (threadIdx.x >> 5); // 0..511
    int m0 = (gw >> 2) * 16;             // sample tile
    int n0 = (gw & 3) * 16;              // out tile
    int ml = threadIdx.x & 15;
    int hi = (threadIdx.x >> 4) & 1;
    int kb = hi * 2;

    const float* ap0 = yflat + (size_t)(m0 + ml) * KFC + kb;
    const float* bp0 = fcw   + (size_t)(n0 + ml) * KFC + kb;
    v8f acc = {};
#pragma unroll 4
    for (int k0 = 0; k0 < KFC; k0 += 4) {
        v2f a  = *(const v2f*)(ap0 + k0);
        v2f bf = *(const v2f*)(bp0 + k0);
        acc = wmma4(a, bf, acc);
    }
#pragma unroll
    for (int v = 0; v < 8; ++v) {
        int s = m0 + v + hi * 8;
        int o = n0 + ml;
        xp[(size_t)s * NC + o] = acc[v] + fcb[o];
    }
}

// ---------------------------------------------------------------
// Kernel 4: attention per batch. scores GEMM -> softmax -> probs GEMM.
// One block (8 waves) per batch. 64KB LDS score/prob matrix.
// ---------------------------------------------------------------
__global__ void k_attn(const float* __restrict__ xp,   // (B,64,64)
                       const float* __restrict__ xf,   // (B,64,256)
                       const float* __restrict__ xfT,  // (B,256,64)
                       float* __restrict__ g) {        // (B,4096)
    __shared__ float S[NLANES][NRP];
    int b = blockIdx.x;
    int tid = threadIdx.x;
    int wave = tid >> 5;
    int ml = tid & 15;
    int hi = (tid >> 4) & 1;
    int kb = hi * 2;
    const float* xpb  = xp  + (size_t)b * NLANES * NC;
    const float* xfb  = xf  + (size_t)b * NC * NRP;
    const float* xfTb = xfT + (size_t)b * NRP * NC;

    // scores(64x256) = x_p(64x64) @ x_f(64x256) ; 4x16 = 64 tiles
    for (int tile = wave; tile < 64; tile += 8) {
        int m0 = (tile >> 4) * 16;
        int n0 = (tile & 15) * 16;
        const float* ap0 = xpb + (size_t)(m0 + ml) * NC + kb;
        v8f acc = {};
#pragma unroll 4
        for (int k0 = 0; k0 < NC; k0 += 4) {
            v2f a = *(const v2f*)(ap0 + k0);
            v2f bf;
            bf.x = xfb[(size_t)(k0 + kb)     * NRP + n0 + ml];
            bf.y = xfb[(size_t)(k0 + kb + 1) * NRP + n0 + ml];
            acc = wmma4(a, bf, acc);
        }
#pragma unroll
        for (int v = 0; v < 8; ++v) S[m0 + v + hi * 8][n0 + ml] = acc[v];
    }
    __syncthreads();

    // row-wise softmax over first 250 cols, scale 1/sqrt(C)=0.125
    for (int row = tid; row < NLANES; row += blockDim.x) {
        float mx = -1e30f;
        for (int r = 0; r < NR; ++r) mx = fmaxf(mx, S[row][r]);
        float sum = 0.0f;
        for (int r = 0; r < NR; ++r) {
            float e = __expf((S[row][r] - mx) * 0.125f);
            S[row][r] = e;
            sum += e;
        }
        float inv = 1.0f / sum;
        for (int r = 0; r < NR; ++r) S[row][r] *= inv;
        for (int r = NR; r < NRP; ++r) S[row][r] = 0.0f;   // mask pads
    }
    __syncthreads();

    // g(64x64) = P(64x256) @ x_f^T(256x64) ; 4x4 = 16 tiles
    for (int tile = wave; tile < 16; tile += 8) {
        int m0 = (tile >> 2) * 16;
        int n0 = (tile & 3) * 16;
        v8f acc = {};
#pragma unroll 4
        for (int k0 = 0; k0 < NRP; k0 += 4) {
            v2f a = *(const v2f*)(&S[m0 + ml][k0 + kb]);   // ds_load_b64
            v2f bf;
            bf.x = xfTb[(size_t)(k0 + kb)     * NC + n0 + ml];
            bf.y = xfTb[(size_t)(k0 + kb + 1) * NC + n0 + ml];
            acc = wmma4(a, bf, acc);
        }
#pragma unroll
        for (int v = 0; v < 8; ++v)
            g[(size_t)b * KG + (m0 + v + hi * 8) * NC + n0 + ml] = acc[v];
    }
}

// ---------------------------------------------------------------
// Kernel 5: gr(32x64) = relu( G(32x4096) @ conv1x1_w^T + b )
// One block, 8 waves = 2 M-tiles x 4 N-tiles, K loop 4096.
// ---------------------------------------------------------------
__global__ void k_conv1x1(const float* __restrict__ g,     // (32,4096)
                          const float* __restrict__ w,     // (64,4096)
                          const float* __restrict__ bias,  // (64)
                          float* __restrict__ gr) {        // (32,64)
    int wave = threadIdx.x >> 5;
    int m0 = (wave >> 2) * 16;           // batch tile: 0,16
    int n0 = (wave & 3) * 16;
    int ml = threadIdx.x & 15;
    int hi = (threadIdx.x >> 4) & 1;
    int kb = hi * 2;

    const float* ap0 = g + (size_t)(m0 + ml) * KG + kb;
    const float* bp0 = w + (size_t)(n0 + ml) * KG + kb;
    v8f acc = {};
#pragma unroll 4
    for (int k0 = 0; k0 < KG; k0 += 4) {
        v2f a  = *(const v2f*)(ap0 + k0);
        v2f bf = *(const v2f*)(bp0 + k0);
        acc = wmma4(a, bf, acc);
    }
#pragma unroll
    for (int v = 0; v < 8; ++v) {
        int bb = m0 + v + hi * 8;
        int o  = n0 + ml;
        gr[bb * NC + o] = fmaxf(acc[v] + bias[o], 0.0f);
    }
}

// ---------------------------------------------------------------
// Kernel 6: out[b][j][p] = gr[b][j] + prior[b][j][p]
// ---------------------------------------------------------------
__global__ void k_out(const float* __restrict__ gr,
                      const float* __restrict__ prior,
                      float* __restrict__ out) {
    int i = blockIdx.x * blockDim.x + threadIdx.x;
    const int total = NB * NLANES * PE_;
    if (i < total) {
        int j = (i / PE_) % NLANES;
        int b = i / (PE_ * NLANES);
        out[i] = gr[b * NC + j] + prior[i];
    }
}

// ---------------------------------------------------------------
extern "C" void kernel_launch(void* const* d_in, const int* in_sizes, int n_in,
                              void* d_out, int out_size, void* d_ws, size_t ws_size,
                              hipStream_t stream) {
    const float* feat  = (const float*)d_in[0];   // (32,64,80,200)
    const float* prior = (const float*)d_in[1];   // (32,64,78)
    const float* c1w   = (const float*)d_in[2];   // (64,64,5)
    const float* c1b   = (const float*)d_in[3];   // (64)
    const float* fcw   = (const float*)d_in[4];   // (64,2048)
    const float* fcb   = (const float*)d_in[5];   // (64)
    const float* cw    = (const float*)d_in[6];   // (64,4096)
    const float* cb    = (const float*)d_in[7];   // (64)
    float* out = (float*)d_out;

    float* ws   = (float*)d_ws;
    float* xf    = ws;                               // 32*64*256   = 524288
    float* xfT   = xf    + (size_t)NB * NC * NRP;    // 524288
    float* yflat = xfT   + (size_t)NB * NRP * NC;    // 2048*2048   = 4194304
    float* xp    = yflat + (size_t)NB * NLANES * KFC;// 2048*64     = 131072
    float* g     = xp    + (size_t)NB * NLANES * NC; // 32*4096     = 131072
    float* gr    = g     + (size_t)NB * KG;          // 32*64       = 2048
    float* wr    = gr    + (size_t)NB * NC;          // 5*64*64     = 20480

    k_prep_w  <<<20,           256, 0, stream>>>(c1w, wr);
    k_pool    <<<NB * NC,      256, 0, stream>>>(feat, xf, xfT);
    k_roi_conv<<<NB * NLANES,  256, 0, stream>>>(feat, prior, wr, c1b, yflat);
    k_fc      <<<64,           256, 0, stream>>>(yflat, fcw, fcb, xp);
    k_attn    <<<NB,           256, 0, stream>>>(xp, xf, xfT, g);
    k_conv1x1 <<<1,            256, 0, stream>>>(g, cw, cb, gr);
    int total = NB * NLANES * PE_;
    k_out     <<<(total + 255) / 256, 256, 0, stream>>>(gr, prior, out);
}